// GCNBottleneck_26723286516384
// MI455X (gfx1250) — compile-verified
//
#include <hip/hip_runtime.h>
#include <hip/hip_bf16.h>

typedef __attribute__((ext_vector_type(2))) float v2f;
typedef __attribute__((ext_vector_type(8))) float v8f;

#define N_NODES 100000
#define N_EDGES 3200000
#define N_GRAPHS 512
#define SCAN_BLOCKS 391   // ceil(100000/256)

// ---------------------------------------------------------------------------
__global__ void zero_i32(int* __restrict__ p, int n) {
    int i = blockIdx.x * blockDim.x + threadIdx.x;
    if (i < n) p[i] = 0;
}
__global__ void zero_f32(float* __restrict__ p, int n) {
    int i = blockIdx.x * blockDim.x + threadIdx.x;
    if (i < n) p[i] = 0.0f;
}

// ---------------------------------------------------------------------------
// CSR build step 1: in-degree count per destination node
// ---------------------------------------------------------------------------
__global__ void count_kernel(const int* __restrict__ dst, int* __restrict__ cnt, int n_edges) {
    int e = blockIdx.x * blockDim.x + threadIdx.x;
    if (e < n_edges) atomicAdd(&cnt[dst[e]], 1);
}

// dinv[n] = rsqrt(cnt[n] + 1)  (+1 = self loop; always > 0)
__global__ void dinv_kernel(const int* __restrict__ cnt, float* __restrict__ dinv, int n) {
    int i = blockIdx.x * blockDim.x + threadIdx.x;
    if (i < n) dinv[i] = rsqrtf((float)cnt[i] + 1.0f);
}

// ---------------------------------------------------------------------------
// 3-phase exclusive prefix scan (counts -> row starts), LDS Hillis-Steele
// ---------------------------------------------------------------------------
__global__ void scan_block(const int* __restrict__ in, int* __restrict__ out,
                           int* __restrict__ bsums, int n) {
    __shared__ int tmp[256];
    int i = blockIdx.x * 256 + threadIdx.x;
    int v = (i < n) ? in[i] : 0;
    tmp[threadIdx.x] = v;
    __syncthreads();
    for (int off = 1; off < 256; off <<= 1) {
        int t = (threadIdx.x >= off) ? tmp[threadIdx.x - off] : 0;
        __syncthreads();
        tmp[threadIdx.x] += t;
        __syncthreads();
    }
    int incl = tmp[threadIdx.x];
    if (i < n) out[i] = incl - v;                      // exclusive
    if (threadIdx.x == 255) bsums[blockIdx.x] = incl;  // block total
}

__global__ void scan_sums(int* __restrict__ bsums, int nb) {
    __shared__ int tmp[512];
    int v = (threadIdx.x < nb) ? bsums[threadIdx.x] : 0;
    tmp[threadIdx.x] = v;
    __syncthreads();
    for (int off = 1; off < 512; off <<= 1) {
        int t = (threadIdx.x >= off) ? tmp[threadIdx.x - off] : 0;
        __syncthreads();
        tmp[threadIdx.x] += t;
        __syncthreads();
    }
    if (threadIdx.x < nb) bsums[threadIdx.x] = tmp[threadIdx.x] - v;  // exclusive
}

__global__ void scan_add(int* __restrict__ out, const int* __restrict__ bsums, int n) {
    int i = blockIdx.x * 256 + threadIdx.x;
    if (i < n) out[i] += bsums[blockIdx.x];
}

// ---------------------------------------------------------------------------
// CSR build step 2: place each edge's source under its destination's row
// ---------------------------------------------------------------------------
__global__ void place_kernel(const int* __restrict__ src, const int* __restrict__ dst,
                             const int* __restrict__ starts, int* __restrict__ fill,
                             int* __restrict__ srcs_sorted, int n_edges) {
    int e = blockIdx.x * blockDim.x + threadIdx.x;
    if (e >= n_edges) return;
    int d = dst[e];
    int pos = starts[d] + atomicAdd(&fill[d], 1);
    srcs_sorted[pos] = src[e];
}

// ---------------------------------------------------------------------------
// HS[N,32] = (X[N,K] @ W[K,32]) * dinv[row]  using V_WMMA_F32_16X16X4_F32.
// One wave: 16 rows x 32 cols (two 16x16 tiles), K/4 k-steps.
// f32 A 16x4: lanes 0-15 -> M=lane, {v0:K=0, v1:K=1}; lanes 16-31 -> K=2,3.
// C/D: VGPR v -> M = v + 8*(lane>=16), N = lane&15.
// ---------------------------------------------------------------------------
template <int K>
__global__ void xform_wmma(const float* __restrict__ X, const float* __restrict__ W,
                           const float* __restrict__ dinv, float* __restrict__ HS, int nrows) {
    int wave = (int)((blockIdx.x * (size_t)blockDim.x + threadIdx.x) >> 5);
    int lane = threadIdx.x & 31;
    int m0 = wave * 16;
    if (m0 >= nrows) return;  // wave-uniform: EXEC stays all-1s for WMMA

    int mh = lane & 15;
    int kh = (lane >> 4) << 1;

    v8f acc0 = {};
    v8f acc1 = {};
    const float* xrow = X + (size_t)(m0 + mh) * K;

#pragma unroll
    for (int kb = 0; kb < K; kb += 4) {
        v2f a, b0, b1;
        a.x = xrow[kb + kh];
        a.y = xrow[kb + kh + 1];
        b0.x = W[(kb + kh) * 32 + mh];
        b0.y = W[(kb + kh + 1) * 32 + mh];
        b1.x = W[(kb + kh) * 32 + 16 + mh];
        b1.y = W[(kb + kh + 1) * 32 + 16 + mh];
        acc0 = __builtin_amdgcn_wmma_f32_16x16x4_f32(false, a, false, b0,
                                                     (short)0, acc0, false, false);
        acc1 = __builtin_amdgcn_wmma_f32_16x16x4_f32(false, a, false, b1,
                                                     (short)0, acc1, false, false);
    }

    int rbase = m0 + ((lane >> 4) << 3);
    float* hout = HS + (size_t)rbase * 32 + mh;
#pragma unroll
    for (int v = 0; v < 8; v++) {
        float dv = dinv[rbase + v];
        hout[(size_t)v * 32]      = acc0[v] * dv;
        hout[(size_t)v * 32 + 16] = acc1[v] * dv;
    }
}

// ---------------------------------------------------------------------------
// Atomic-free aggregation: one wave per destination node, lane = channel.
// act[d,c] = relu( (sum_{s in N(d)} hs[s,c] + hs[d,c]) * dinv[d] + bias[c] )
// (hs = h*dinv[s], so this equals sum h[s]*dinv[s]*dinv[d] + h[d]*dinv[d]^2.)
// Each in-edge row is one coalesced 128B L2-resident load.
// ---------------------------------------------------------------------------
__global__ void aggregate_kernel(const int* __restrict__ starts, const int* __restrict__ cnts,
                                 const int* __restrict__ srcs, const float* __restrict__ hs,
                                 const float* __restrict__ dinv, const float* __restrict__ bias,
                                 float* __restrict__ act, int n_nodes) {
    int d = (int)((blockIdx.x * (size_t)blockDim.x + threadIdx.x) >> 5);
    int c = threadIdx.x & 31;
    if (d >= n_nodes) return;

    int beg = starts[d];
    int cnt = cnts[d];
    float acc = hs[(size_t)d * 32 + c];  // self loop

    int j = 0;
    for (; j + 1 < cnt; j += 2) {        // 2-deep software pipeline
        int s0 = srcs[beg + j];
        int s1 = srcs[beg + j + 1];
        acc += hs[(size_t)s0 * 32 + c];
        acc += hs[(size_t)s1 * 32 + c];
    }
    if (j < cnt) acc += hs[(size_t)srcs[beg + j] * 32 + c];

    float v = acc * dinv[d] + bias[c];
    act[(size_t)d * 32 + c] = fmaxf(v, 0.0f);
}

// ---------------------------------------------------------------------------
// Pool accumulate + classifier
// ---------------------------------------------------------------------------
__global__ void pool_acc(const float* __restrict__ act, const int* __restrict__ batch,
                         float* __restrict__ sums, float* __restrict__ cnt, int n_nodes) {
    long long idx = (long long)blockIdx.x * blockDim.x + threadIdx.x;
    int n = (int)(idx >> 5);
    int c = (int)(idx & 31);
    if (n >= n_nodes) return;
    int g = batch[n];
    atomicAdd(&sums[(size_t)g * 32 + c], act[(size_t)n * 32 + c]);
    if (c == 0) atomicAdd(&cnt[g], 1.0f);
}

__global__ void classify(const float* __restrict__ sums, const float* __restrict__ cnt,
                         const float* __restrict__ Wc, const float* __restrict__ bc,
                         float* __restrict__ out) {
    int i = blockIdx.x * blockDim.x + threadIdx.x;
    if (i >= N_GRAPHS * 2) return;
    int g = i >> 1;
    int c = i & 1;
    float inv = 1.0f / fmaxf(cnt[g], 1.0f);
    float acc = bc[c];
#pragma unroll
    for (int k = 0; k < 32; k++) acc += sums[(size_t)g * 32 + k] * inv * Wc[k * 2 + c];
    out[i] = acc;
}

// ---------------------------------------------------------------------------
extern "C" void kernel_launch(void* const* d_in, const int* in_sizes, int n_in,
                              void* d_out, int out_size, void* d_ws, size_t ws_size,
                              hipStream_t stream) {
    const float* x   = (const float*)d_in[0];
    const int*   ei  = (const int*)d_in[1];  // (2, E) flat
    const int*   src = ei;
    const int*   dst = ei + N_EDGES;
    const int*   bat = (const int*)d_in[2];
    const float* W1  = (const float*)d_in[3];
    const float* b1  = (const float*)d_in[4];
    const float* W2  = (const float*)d_in[5];
    const float* b2  = (const float*)d_in[6];
    const float* Wc  = (const float*)d_in[7];
    const float* bc  = (const float*)d_in[8];
    float* out = (float*)d_out;

    // workspace layout
    int*   cnt_i  = (int*)d_ws;                    // 100352 (padded)
    int*   starts = cnt_i + 100352;                // 100352
    int*   fill   = starts + 100352;               // 100352
    int*   bsums  = fill + 100352;                 // 512
    int*   srcs_s = bsums + 512;                   // N_EDGES
    float* dinv   = (float*)(srcs_s + N_EDGES);    // 100352
    float* hs     = dinv + 100352;                 // N*32
    float* act    = hs + (size_t)N_NODES * 32;     // N*32
    float* sums   = act + (size_t)N_NODES * 32;    // 512*32
    float* gcnt   = sums + (size_t)N_GRAPHS * 32;  // 512

    const int B = 256;
    const int n_blocks  = (N_NODES + B - 1) / B;
    const int e_blocks  = (N_EDGES + B - 1) / B;
    const int nc_blocks = (int)(((long long)N_NODES * 32 + B - 1) / B);
    const int x_blocks  = (N_NODES / 16 * 32 + B - 1) / B;  // 6250 waves

    // --- CSR build + dinv (shared by both layers) ---
    zero_i32<<<n_blocks, B, 0, stream>>>(cnt_i, N_NODES);
    count_kernel<<<e_blocks, B, 0, stream>>>(dst, cnt_i, N_EDGES);
    dinv_kernel<<<n_blocks, B, 0, stream>>>(cnt_i, dinv, N_NODES);
    scan_block<<<SCAN_BLOCKS, 256, 0, stream>>>(cnt_i, starts, bsums, N_NODES);
    scan_sums<<<1, 512, 0, stream>>>(bsums, SCAN_BLOCKS);
    scan_add<<<SCAN_BLOCKS, 256, 0, stream>>>(starts, bsums, N_NODES);
    zero_i32<<<n_blocks, B, 0, stream>>>(fill, N_NODES);
    place_kernel<<<e_blocks, B, 0, stream>>>(src, dst, starts, fill, srcs_s, N_EDGES);

    // --- layer 1: hs = (x@W1)*dinv ; act = relu(gather*dinv + b1) ---
    xform_wmma<8><<<x_blocks, B, 0, stream>>>(x, W1, dinv, hs, N_NODES);
    aggregate_kernel<<<nc_blocks, B, 0, stream>>>(starts, cnt_i, srcs_s, hs, dinv, b1, act, N_NODES);

    // --- layer 2 ---
    xform_wmma<32><<<x_blocks, B, 0, stream>>>(act, W2, dinv, hs, N_NODES);
    aggregate_kernel<<<nc_blocks, B, 0, stream>>>(starts, cnt_i, srcs_s, hs, dinv, b2, act, N_NODES);

    // --- pool + classify ---
    zero_f32<<<(N_GRAPHS * 33 + B - 1) / B, B, 0, stream>>>(sums, N_GRAPHS * 33);
    pool_acc<<<nc_blocks, B, 0, stream>>>(act, bat, sums, gcnt, N_NODES);
    classify<<<(N_GRAPHS * 2 + B - 1) / B, B, 0, stream>>>(sums, gcnt, Wc, bc, out);
}